// KalmanFilter_19731079757905
// MI455X (gfx1250) — compile-verified
//
#include <hip/hip_runtime.h>
#include <math.h>

typedef float v8f __attribute__((ext_vector_type(8)));
typedef float v2f __attribute__((ext_vector_type(2)));

#define N_OBS   262144
#define CHUNK   512
#define NCHUNK  (N_OBS / CHUNK)
#define LOG2PI  1.8378770664093453f

__device__ __forceinline__ float rcp_f(float x) { return __builtin_amdgcn_rcpf(x); }

// Broadcast lane i's value to all lanes via v_readlane (no LDS round-trip).
__device__ __forceinline__ float rdlane(float x, int i) {
    unsigned u = __builtin_bit_cast(unsigned, x);
    unsigned r = __builtin_amdgcn_readlane(u, (unsigned)i);
    return __builtin_bit_cast(float, r);
}

// ---------------------------------------------------------------------------
// Kernel 1: extract diag(F), diag(Q) into compact stream pk[n][16] = {f[8],q[8]}
// Also zero the output accumulator.
// ---------------------------------------------------------------------------
__global__ void k_pack(const float* __restrict__ F, const float* __restrict__ Q,
                       float* __restrict__ pk, float* __restrict__ out) {
    if (blockIdx.x == 0 && threadIdx.x == 0) out[0] = 0.0f;
    int idx = blockIdx.x * blockDim.x + threadIdx.x;
    const int total = N_OBS * 8;
    const int stride = gridDim.x * blockDim.x;
    for (; idx < total; idx += stride) {
        int n = idx >> 3, i = idx & 7;
        pk[n * 16 + i]     = F[n * 64 + i * 9];
        pk[n * 16 + 8 + i] = Q[n * 64 + i * 9];
    }
}

// ---------------------------------------------------------------------------
// Kernel 2: per-chunk homogeneous covariance transfer matrix T (16x16).
// Two chunks per wave (one per 16-lane half); lane owns one column of T.
// Step map on [U;V]: U <- f.*U + (q./f).*V ; V <- V./f ; V += h*(h^T U)/r
// Scale-renormalized every 128 steps (P = U V^-1 is scale invariant);
// final T leaves with max|elem| == 1, which bounds phase-2 growth.
// ---------------------------------------------------------------------------
__global__ void __launch_bounds__(32)
k_chunkT(const float* __restrict__ pk, const float* __restrict__ H,
         const float* __restrict__ errors, const float* __restrict__ nu_p,
         float* __restrict__ Tws) {
    const int lane  = threadIdx.x;
    const int col   = lane & 15;
    const int chunk = blockIdx.x * 2 + (lane >> 4);
    const float nu  = nu_p[0];
    float h[8];
#pragma unroll
    for (int i = 0; i < 8; ++i) h[i] = H[i];
    float T[16];
#pragma unroll
    for (int r = 0; r < 16; ++r) T[r] = (r == col) ? 1.0f : 0.0f;
    const int base = chunk * CHUNK;
    for (int k = 0; k < CHUNK; ++k) {
        const int g = base + k;
        const float4* p4 = (const float4*)(pk + g * 16);
        float4 f0 = p4[0], f1 = p4[1], q0 = p4[2], q1 = p4[3];
        float f[8] = {f0.x, f0.y, f0.z, f0.w, f1.x, f1.y, f1.z, f1.w};
        float q[8] = {q0.x, q0.y, q0.z, q0.w, q1.x, q1.y, q1.z, q1.w};
        float e = errors[g];
        float rinv = rcp_f(nu * e * e);
#pragma unroll
        for (int i = 0; i < 8; ++i) {
            float fi = f[i], fir = rcp_f(fi);
            T[i]     = fi * T[i] + (q[i] * fir) * T[8 + i];
            T[8 + i] = fir * T[8 + i];
        }
        float dot = 0.f;
#pragma unroll
        for (int i = 0; i < 8; ++i) dot += h[i] * T[i];
        dot *= rinv;
#pragma unroll
        for (int i = 0; i < 8; ++i) T[8 + i] += h[i] * dot;
        if ((k & 127) == 127) {
            float m = 0.f;
#pragma unroll
            for (int r = 0; r < 16; ++r) m = fmaxf(m, fabsf(T[r]));
#pragma unroll
            for (int off = 1; off < 16; off <<= 1) m = fmaxf(m, __shfl_xor(m, off));
            float s = rcp_f(m);
#pragma unroll
            for (int r = 0; r < 16; ++r) T[r] *= s;
        }
    }
#pragma unroll
    for (int r = 0; r < 16; ++r) Tws[chunk * 256 + r * 16 + col] = T[r];
}

// ---------------------------------------------------------------------------
// Kernel 3: sequential inter-chunk prefix over the 16x16 transfer matrices.
// state (16x16, cols 0..7 = [U;V], cols 8..15 stay zero) is updated as
// state <- T_c @ state using 4 chained V_WMMA_F32_16X16X4_F32 (K=16).
// The exclusive prefix (chunk starting [U;V]) is stored for phase 3.
// Renormalize every 8 chunks (per-chunk growth <= 16x since max|T| = 1).
// Single wave: LDS ops are in-order, no barriers needed for the bounce.
// ---------------------------------------------------------------------------
__global__ void __launch_bounds__(32)
k_prefix(const float* __restrict__ Tws, const float* __restrict__ P0,
         float* __restrict__ stws) {
    __shared__ float lds[256];
    const int lane = threadIdx.x;
    const int half = lane >> 4;      // 0 or 1
    const int n    = lane & 15;      // column index
    v8f st;
#pragma unroll
    for (int r = 0; r < 8; ++r) {
        int M = r + 8 * half;
        float v = 0.f;
        if (n < 8) {
            if (M < 8) v = P0[M * 8 + n];              // U0 = P0
            else       v = (M - 8 == n) ? 1.f : 0.f;   // V0 = I
        }
        st[r] = v;
    }
    for (int c = 0; c < NCHUNK; ++c) {
        // store exclusive-prefix state (chunk c's starting [U;V])
#pragma unroll
        for (int r = 0; r < 8; ++r) {
            int M = r + 8 * half;
            stws[c * 256 + M * 16 + n] = st[r];
        }
        // A operand: T_c in 16x4 f32 A-tile layout
        v2f A0, A1, A2, A3;
        {
            const float* Tc = Tws + c * 256 + n * 16;
            int ko = 2 * half;
            A0[0] = Tc[0 + ko];  A0[1] = Tc[1 + ko];
            A1[0] = Tc[4 + ko];  A1[1] = Tc[5 + ko];
            A2[0] = Tc[8 + ko];  A2[1] = Tc[9 + ko];
            A3[0] = Tc[12 + ko]; A3[1] = Tc[13 + ko];
        }
        // convert state (C/D layout) -> B-tile layout via LDS bounce
        // (single wave: DS ops retire in-order; compiler tracks the RAW dep)
#pragma unroll
        for (int r = 0; r < 8; ++r) {
            int M = r + 8 * half;
            lds[M * 16 + n] = st[r];
        }
        v2f B0, B1, B2, B3;
        {
            int ko = 2 * half;
            B0[0] = lds[(0 + ko) * 16 + n];  B0[1] = lds[(1 + ko) * 16 + n];
            B1[0] = lds[(4 + ko) * 16 + n];  B1[1] = lds[(5 + ko) * 16 + n];
            B2[0] = lds[(8 + ko) * 16 + n];  B2[1] = lds[(9 + ko) * 16 + n];
            B3[0] = lds[(12 + ko) * 16 + n]; B3[1] = lds[(13 + ko) * 16 + n];
        }
        v8f d = {0.f, 0.f, 0.f, 0.f, 0.f, 0.f, 0.f, 0.f};
        d = __builtin_amdgcn_wmma_f32_16x16x4_f32(false, A0, false, B0, (short)0, d, false, false);
        d = __builtin_amdgcn_wmma_f32_16x16x4_f32(false, A1, false, B1, (short)0, d, false, false);
        d = __builtin_amdgcn_wmma_f32_16x16x4_f32(false, A2, false, B2, (short)0, d, false, false);
        d = __builtin_amdgcn_wmma_f32_16x16x4_f32(false, A3, false, B3, (short)0, d, false, false);
        if ((c & 7) == 7) {
            // renormalize (zero columns stay zero; V block keeps max > 0)
            float m = 0.f;
#pragma unroll
            for (int r = 0; r < 8; ++r) m = fmaxf(m, fabsf(d[r]));
#pragma unroll
            for (int off = 1; off < 32; off <<= 1) m = fmaxf(m, __shfl_xor(m, off));
            float s = rcp_f(m);
#pragma unroll
            for (int r = 0; r < 8; ++r) st[r] = d[r] * s;
        } else {
#pragma unroll
            for (int r = 0; r < 8; ++r) st[r] = d[r];
        }
    }
}

// ---------------------------------------------------------------------------
// Kernel 4: per-chunk exact covariance scan.
// Recover P_start = U V^-1 (Gauss-Jordan on V^T P^T = U^T, one row per lane),
// then run 512 exact steps. Lanes 0-7: columns of P; lanes 8-15: columns of
// the chunk mean-transfer M; lanes 16-31: the affine offset c (replicated).
// Emits per-step S and K, and per-chunk (M, c). All cross-lane traffic is
// fixed-lane broadcast -> v_readlane (no LDS permutes on the step chain).
// ---------------------------------------------------------------------------
__global__ void __launch_bounds__(32)
k_scanP(const float* __restrict__ pk, const float* __restrict__ H,
        const float* __restrict__ values, const float* __restrict__ errors,
        const float* __restrict__ mu_p, const float* __restrict__ nu_p,
        const float* __restrict__ stws,
        float* __restrict__ Sws, float* __restrict__ Kws,
        float* __restrict__ Mcws) {
    const int lane  = threadIdx.x;
    const int chunk = blockIdx.x;
    const int j     = lane & 7;
    const int role  = (lane < 8) ? 0 : ((lane < 16) ? 1 : 2);
    const float mu  = mu_p[0];
    const float nu  = nu_p[0];
    float h[8];
#pragma unroll
    for (int i = 0; i < 8; ++i) h[i] = H[i];

    // ---- P_start = U V^-1 : solve V^T P^T = U^T, lane j holds row j ----
    float a[8], b[8];
#pragma unroll
    for (int r = 0; r < 8; ++r) {
        a[r] = stws[chunk * 256 + (8 + r) * 16 + j];  // (V^T) row j
        b[r] = stws[chunk * 256 + r * 16 + j];        // (U^T) row j
    }
#pragma unroll
    for (int kk = 0; kk < 8; ++kk) {
        float piv  = rdlane(a[kk], kk);
        float pinv = rcp_f(piv);
        float ra[8], rb[8];
#pragma unroll
        for (int r = 0; r < 8; ++r) {
            ra[r] = rdlane(a[r], kk) * pinv;
            rb[r] = rdlane(b[r], kk) * pinv;
        }
        float fm  = a[kk];
        bool  isp = (j == kk);
#pragma unroll
        for (int r = 0; r < 8; ++r) {
            a[r] = isp ? ra[r] : (a[r] - fm * ra[r]);
            b[r] = isp ? rb[r] : (b[r] - fm * rb[r]);
        }
    }
    // lane j (role 0) now holds P column j in b[]
    float col[8];
#pragma unroll
    for (int r = 0; r < 8; ++r) {
        float v = 0.f;
        if (role == 0) v = b[r];
        else if (role == 1) v = (r == j) ? 1.f : 0.f;   // M = I
        col[r] = v;                                      // c = 0
    }

    const int base = chunk * CHUNK;
    for (int k = 0; k < CHUNK; ++k) {
        const int g = base + k;
        const float4* p4 = (const float4*)(pk + g * 16);
        float4 f0 = p4[0], f1 = p4[1], q0 = p4[2], q1 = p4[3];
        float f[8] = {f0.x, f0.y, f0.z, f0.w, f1.x, f1.y, f1.z, f1.w};
        float q[8] = {q0.x, q0.y, q0.z, q0.w, q1.x, q1.y, q1.z, q1.w};
        float z = values[g], e = errors[g];
        float rr = nu * e * e;
        float f_l = 0.f;
#pragma unroll
        for (int i = 0; i < 8; ++i) f_l = (j == i) ? f[i] : f_l;
        // predict
        if (role == 0) {
#pragma unroll
            for (int i = 0; i < 8; ++i)
                col[i] = col[i] * f[i] * f_l + ((i == j) ? q[i] : 0.f);
        } else {
#pragma unroll
            for (int i = 0; i < 8; ++i) col[i] *= f[i];
        }
        // per-column dot with h: for P lanes this is w_j = (P' h)_j (symmetry)
        float dt = 0.f;
#pragma unroll
        for (int i = 0; i < 8; ++i) dt += h[i] * col[i];
        // S = nu e^2 + h^T P' h  via readlane broadcasts of w
        float hPh = 0.f;
#pragma unroll
        for (int i = 0; i < 8; ++i) hPh += h[i] * rdlane(dt, i);
        float S    = rr + hPh;
        float Sinv = rcp_f(S);
        float Kj   = dt * Sinv;   // Kalman gain component (valid on P lanes)
        float ki[8];
#pragma unroll
        for (int i = 0; i < 8; ++i) ki[i] = rdlane(Kj, i);
        if (role == 0) {
            float sk = S * Kj;
#pragma unroll
            for (int i = 0; i < 8; ++i) col[i] -= sk * ki[i];   // P -= S K K^T
            if (lane == 0) Sws[g] = S;
            Kws[g * 8 + lane] = Kj;
        } else {
            // (I - K h^T) applied to F*col
#pragma unroll
            for (int i = 0; i < 8; ++i) col[i] -= ki[i] * dt;
            if (role == 2) {
                float ymu = z - mu;
#pragma unroll
                for (int i = 0; i < 8; ++i) col[i] += ki[i] * ymu;
            }
        }
    }
    if (role == 1) {
#pragma unroll
        for (int r = 0; r < 8; ++r) Mcws[chunk * 72 + r * 8 + j] = col[r];
    } else if (lane == 16) {
#pragma unroll
        for (int r = 0; r < 8; ++r) Mcws[chunk * 72 + 64 + r] = col[r];
    }
}

// ---------------------------------------------------------------------------
// Kernel 5: sequential prefix of per-chunk affine mean maps: X <- M_c X + c_c.
// Lane j holds row j of M_c; X kept replicated per lane via readlane.
// ---------------------------------------------------------------------------
__global__ void __launch_bounds__(32)
k_meanPrefix(const float* __restrict__ Mcws, const float* __restrict__ X0,
             float* __restrict__ Xsws) {
    const int lane = threadIdx.x;
    const int jj   = lane & 7;
    float X[8];
#pragma unroll
    for (int i = 0; i < 8; ++i) X[i] = X0[i];
    for (int c = 0; c < NCHUNK; ++c) {
        if (lane < 8) {
            float xv = 0.f;
#pragma unroll
            for (int i = 0; i < 8; ++i) xv = (jj == i) ? X[i] : xv;
            Xsws[c * 8 + lane] = xv;                 // exclusive prefix
        }
        float y = Mcws[c * 72 + 64 + jj];
#pragma unroll
        for (int i = 0; i < 8; ++i) y += Mcws[c * 72 + jj * 8 + i] * X[i];
#pragma unroll
        for (int i = 0; i < 8; ++i) X[i] = rdlane(y, i);
    }
}

// ---------------------------------------------------------------------------
// Kernel 6: per-chunk likelihood pass. Lane j carries X_j; uses stored S, K.
// Innovation reduce done with readlane broadcasts (no LDS permutes).
// ---------------------------------------------------------------------------
__global__ void __launch_bounds__(32)
k_ll(const float* __restrict__ pk, const float* __restrict__ H,
     const float* __restrict__ values, const float* __restrict__ mu_p,
     const float* __restrict__ Sws, const float* __restrict__ Kws,
     const float* __restrict__ Xsws, float* __restrict__ out) {
    const int lane  = threadIdx.x;
    const int chunk = blockIdx.x;
    const float mu  = mu_p[0];
    float h[8];
#pragma unroll
    for (int i = 0; i < 8; ++i) h[i] = H[i];
    float X   = (lane < 8) ? Xsws[chunk * 8 + lane] : 0.f;
    float acc = 0.f;
    const int base = chunk * CHUNK;
    for (int k = 0; k < CHUNK; ++k) {
        const int g  = base + k;
        float fl = (lane < 8) ? pk[g * 16 + lane] : 0.f;
        X *= fl;                                        // predict mean
        float hX = 0.f;
#pragma unroll
        for (int i = 0; i < 8; ++i) hX += h[i] * rdlane(X, i);
        float Y  = values[g] - mu - hX;
        float S  = Sws[g];
        float Kv = (lane < 8) ? Kws[g * 8 + lane] : 0.f;
        X += Kv * Y;
        acc += __logf(S) + Y * Y * rcp_f(S);
    }
    if (lane == 0) atomicAdd(out, -0.5f * (acc + (float)CHUNK * LOG2PI));
}

// ---------------------------------------------------------------------------
extern "C" void kernel_launch(void* const* d_in, const int* in_sizes, int n_in,
                              void* d_out, int out_size, void* d_ws, size_t ws_size,
                              hipStream_t stream) {
    (void)in_sizes; (void)n_in; (void)out_size; (void)ws_size;
    const float* F  = (const float*)d_in[0];
    const float* Q  = (const float*)d_in[1];
    const float* H  = (const float*)d_in[2];
    const float* zv = (const float*)d_in[3];
    const float* ev = (const float*)d_in[4];
    const float* X0 = (const float*)d_in[5];
    const float* P0 = (const float*)d_in[6];
    const float* mu = (const float*)d_in[7];
    const float* nu = (const float*)d_in[8];
    float* out = (float*)d_out;

    float* ws   = (float*)d_ws;
    float* pk   = ws;                                   // N*16
    float* Tws  = pk   + (size_t)N_OBS * 16;            // NCHUNK*256
    float* stws = Tws  + (size_t)NCHUNK * 256;          // NCHUNK*256
    float* Sws  = stws + (size_t)NCHUNK * 256;          // N
    float* Kws  = Sws  + (size_t)N_OBS;                 // N*8
    float* Mcws = Kws  + (size_t)N_OBS * 8;             // NCHUNK*72
    float* Xsws = Mcws + (size_t)NCHUNK * 72;           // NCHUNK*8

    k_pack<<<4096, 256, 0, stream>>>(F, Q, pk, out);
    k_chunkT<<<NCHUNK / 2, 32, 0, stream>>>(pk, H, ev, nu, Tws);
    k_prefix<<<1, 32, 0, stream>>>(Tws, P0, stws);
    k_scanP<<<NCHUNK, 32, 0, stream>>>(pk, H, zv, ev, mu, nu, stws, Sws, Kws, Mcws);
    k_meanPrefix<<<1, 32, 0, stream>>>(Mcws, X0, Xsws);
    k_ll<<<NCHUNK, 32, 0, stream>>>(pk, H, zv, mu, Sws, Kws, Xsws, out);
}